// SelfAttention_49168785605364
// MI455X (gfx1250) — compile-verified
//
#include <hip/hip_runtime.h>

typedef __attribute__((ext_vector_type(2))) float v2f;
typedef __attribute__((ext_vector_type(8))) float v8f;

#define F_DIM  64
#define H_DIM  4
#define FH_DIM 16
#define STRIP  128   // pairs per 64-thread group in the scatter phase

// ---------------------------------------------------------------------------
// Phase 1: per 16-node tile, per head h:
//   Q = X_h * Wq[h], K = X_h * Wk[h], V = X_h * Wv[h]   (all 16x16x16 f32)
// computed as 4 chained V_WMMA_F32_16X16X4_F32 (K = 4 chunks of 4).
// Outputs:
//   s_out[n*H + h]          = (Q[n,:] . K[n,:]) / sqrt(FH)
//   vproj[n*F + h*FH + g]   = V[n,g]
// One wave (32 lanes) per 16-node tile. EXEC all-1s at every WMMA
// (early-exit is wave-uniform; store guards restore full EXEC afterwards).
// ---------------------------------------------------------------------------
__global__ __launch_bounds__(256) void qkv_wmma_kernel(
    const float* __restrict__ x,
    const float* __restrict__ Wq,
    const float* __restrict__ Wk,
    const float* __restrict__ Wv,
    float* __restrict__ s_out,
    float* __restrict__ vproj,
    int n, int ntiles)
{
    const int wave = (int)((blockIdx.x * blockDim.x + threadIdx.x) >> 5);
    if (wave >= ntiles) return;                 // wave-uniform exit
    const int lane = (int)(threadIdx.x & 31);
    const int col  = lane & 15;                 // A-row / B-col / C-col
    const int kh   = lane >> 4;                 // K-half (A/B) and M-half (C/D)
    const int m0   = wave * 16;

    #pragma unroll
    for (int h = 0; h < H_DIM; ++h) {
        v8f qa = {}; v8f ka = {}; v8f va = {};
        #pragma unroll
        for (int c = 0; c < 4; ++c) {
            // ----- A chunk (16x4 f32): VGPRv <- K = c*4 + v + 2*kh, M = col
            const int arow = (m0 + col < n) ? (m0 + col) : (n - 1);
            const int kidx = c * 4 + 2 * kh;
            v2f a;
            a.x = x[(size_t)arow * F_DIM + h * FH_DIM + kidx];
            a.y = x[(size_t)arow * F_DIM + h * FH_DIM + kidx + 1];
            // ----- B chunks (4x16 f32): VGPRv <- K = c*4 + v + 2*kh, N = col
            const size_t wb = (size_t)h * FH_DIM * FH_DIM;
            v2f bq, bk, bv;
            bq.x = Wq[wb + (size_t)kidx       * FH_DIM + col];
            bq.y = Wq[wb + (size_t)(kidx + 1) * FH_DIM + col];
            bk.x = Wk[wb + (size_t)kidx       * FH_DIM + col];
            bk.y = Wk[wb + (size_t)(kidx + 1) * FH_DIM + col];
            bv.x = Wv[wb + (size_t)kidx       * FH_DIM + col];
            bv.y = Wv[wb + (size_t)(kidx + 1) * FH_DIM + col];
            // D = A*B + C   (8 args: neg_a, A, neg_b, B, c_mod, C, reuse_a, reuse_b)
            qa = __builtin_amdgcn_wmma_f32_16x16x4_f32(false, a, false, bq, (short)0, qa, false, false);
            ka = __builtin_amdgcn_wmma_f32_16x16x4_f32(false, a, false, bk, (short)0, ka, false, false);
            va = __builtin_amdgcn_wmma_f32_16x16x4_f32(false, a, false, bv, (short)0, va, false, false);
        }

        // ----- store V tile: C/D VGPRr holds row M = r + 8*kh, column = col
        #pragma unroll
        for (int r = 0; r < 8; ++r) {
            const int node = m0 + r + 8 * kh;
            if (node < n)
                vproj[(size_t)node * F_DIM + h * FH_DIM + col] = va[r];
        }

        // ----- per-row dot(Q,K): reduce across the 16 columns (lanes of each half)
        float d[8];
        #pragma unroll
        for (int r = 0; r < 8; ++r) d[r] = qa[r] * ka[r];
        #pragma unroll
        for (int m = 1; m < 16; m <<= 1) {
            #pragma unroll
            for (int r = 0; r < 8; ++r) d[r] += __shfl_xor(d[r], m, 16);
        }
        // cndmask chain (no dynamic register indexing -> no scratch)
        float sval = d[0];
        #pragma unroll
        for (int r = 1; r < 8; ++r) sval = (col == r) ? d[r] : sval;
        if (col < 8) {
            const int node = m0 + col + 8 * kh;
            if (node < n)
                s_out[(size_t)node * H_DIM + h] = sval * 0.25f;   // 1/sqrt(16)
        }
    }
}

// ---------------------------------------------------------------------------
// Phase 2: edge scatter. 64-thread group = one feature lane per thread,
// walks a contiguous strip of pairs. idx_i is sorted, so accumulate in a
// register while the segment id is unchanged and atomically flush only on
// run boundaries (avg run length = P/N = 16 -> ~16x fewer atomics).
// out[] fits in the 192MB L2, so the remaining atomics stay on-chip.
// ---------------------------------------------------------------------------
__global__ __launch_bounds__(256) void edge_scatter_kernel(
    const float* __restrict__ s,
    const float* __restrict__ vproj,
    const float* __restrict__ phi,
    const int*   __restrict__ idx_i,
    const int*   __restrict__ idx_j,
    const float* __restrict__ mask,
    float* __restrict__ out,
    int P)
{
    const int group = (int)((blockIdx.x * blockDim.x + threadIdx.x) >> 6);
    const int t = (int)(threadIdx.x & 63);   // feature 0..63
    const int h = t >> 4;                    // head
    long p0 = (long)group * STRIP;
    if (p0 >= P) return;                     // wave-uniform
    long p1 = p0 + STRIP; if (p1 > P) p1 = P;

    float acc = 0.0f;
    int cur = -1;
    for (long p = p0; p < p1; ++p) {
        const int   i = idx_i[p];            // same across group -> broadcast load
        const int   j = idx_j[p];
        const float w = mask[p] * phi[p];
        if (i != cur) {                      // wave-uniform branch
            if (cur >= 0) atomicAdd(&out[(size_t)cur * F_DIM + t], acc);
            acc = 0.0f; cur = i;
        }
        const float alpha = s[(size_t)j * H_DIM + h] * w;
        acc = fmaf(alpha, vproj[(size_t)j * F_DIM + t], acc);   // coalesced 256B gather
    }
    if (cur >= 0) atomicAdd(&out[(size_t)cur * F_DIM + t], acc);
}

// ---------------------------------------------------------------------------
extern "C" void kernel_launch(void* const* d_in, const int* in_sizes, int n_in,
                              void* d_out, int out_size, void* d_ws, size_t ws_size,
                              hipStream_t stream) {
    const float* x     = (const float*)d_in[0];
    const float* phi   = (const float*)d_in[1];
    const int*   idx_i = (const int*)  d_in[2];
    const int*   idx_j = (const int*)  d_in[3];
    const float* mask  = (const float*)d_in[4];
    const float* Wq    = (const float*)d_in[5];
    const float* Wk    = (const float*)d_in[6];
    const float* Wv    = (const float*)d_in[7];
    float* out = (float*)d_out;

    const int n = in_sizes[0] / F_DIM;       // 100000
    const int P = in_sizes[1];               // 1600000

    // workspace: s[n*H] then vproj[n*F]  (~27.2 MB total for n=100000)
    float* s  = (float*)d_ws;
    float* vp = s + (size_t)n * H_DIM;

    hipMemsetAsync(out, 0, (size_t)n * F_DIM * sizeof(float), stream);

    const int ntiles = (n + 15) / 16;
    const int wpb = 256 / 32;                // waves per block
    qkv_wmma_kernel<<<dim3((ntiles + wpb - 1) / wpb), 256, 0, stream>>>(
        x, Wq, Wk, Wv, s, vp, n, ntiles);

    const int ngroups = (P + STRIP - 1) / STRIP;
    const int gpb = 256 / 64;                // groups per block
    edge_scatter_kernel<<<dim3((ngroups + gpb - 1) / gpb), 256, 0, stream>>>(
        s, vp, phi, idx_i, idx_j, mask, out, P);
}